// LatentCEDisRNNAgent_47201690583680
// MI455X (gfx1250) — compile-verified
//
#include <hip/hip_runtime.h>
#include <hip/hip_bf16.h>
#include <math.h>

// ---------------------------------------------------------------------------
// Model dims
// ---------------------------------------------------------------------------
#define BS   128
#define AA   64
#define F_IN 128
#define LL   64
#define HH   64
#define NNH  64
#define NACT 16
#define ROWS (BS*AA)          // 8192
#define LOG2PI 1.8378770664093453f

// ---------------------------------------------------------------------------
// Workspace layout (float element offsets)
// ---------------------------------------------------------------------------
#define WS_SUPPORT  0L            // 8192*128
#define WS_GC       1048576L      // 8192*128
#define WS_MU       2097152L      // 8192*64
#define WS_SIGMA    2621440L
#define WS_LATENT   3145728L
#define WS_INFIN    3670016L      // 8192*192
#define WS_INFH     5242880L      // 8192*64
#define WS_INFOUT   5767168L      // 8192*128
#define WS_T1       6815744L      // 8192*64
#define WS_T2       7340032L
#define WS_Z        7864320L
#define WS_FC2W     8388608L      // 8192*1024
#define WS_FC2B     16777216L     // 8192*16
#define WS_X        16908288L     // 8192*64
#define WS_GI       17432576L     // 8192*192
#define WS_GH       19005440L
#define WS_H        20578304L     // 8192*64
#define WS_MICAT    21102592L     // 128*4096
#define WS_DISCAT   21626880L
#define WS_BNMEAN   22151168L     // 64
#define WS_BNINV    22151232L
#define WS_DHMEAN   22151296L     // 64*64
#define WS_DHINV    22155392L
#define WS_MIMIN    22159488L     // 128 each
#define WS_MIMAX    22159616L
#define WS_DIMIN    22159744L
#define WS_DIMAX    22159872L
#define WS_ACC      22160000L     // [ent_sum, kl_sum, clip_sum, norm_sum]

typedef __attribute__((ext_vector_type(2))) float v2f;
typedef __attribute__((ext_vector_type(8))) float v8f;

__device__ __forceinline__ float leakyf(float v) { return v > 0.f ? v : 0.01f * v; }

// ---------------------------------------------------------------------------
// Generic WMMA fp32 GEMM: C[b] = act(A[b] @ B[b] + bias)
//   A: M x K row-major (lda=K).  B: K x N row-major (ldb=N), or if TRANS_B,
//   B is stored N x K (ldb=K) and we compute A @ B^T.
//   MODE: 0 = none, 1 = +bias, 2 = +bias,relu, 3 = relu
//   One wave32 -> one 16x16 C tile via V_WMMA_F32_16X16X4_F32.
// ---------------------------------------------------------------------------
template<int TRANS_B, int MODE>
__global__ __launch_bounds__(128) void gemm_wmma(
    const float* __restrict__ A, const float* __restrict__ B,
    const float* __restrict__ bias, float* __restrict__ C,
    int M, int N, int K, long strideA, long strideB, long strideC)
{
  const int wave = threadIdx.x >> 5;
  const int lane = threadIdx.x & 31;
  const int half = lane >> 4;          // 0: lanes 0-15, 1: lanes 16-31
  const int l15  = lane & 15;
  const int tileN = blockIdx.x * 16;
  const int tileM = (blockIdx.y * 4 + wave) * 16;
  if (tileM >= M) return;

  const float* Ab = A + (long)blockIdx.z * strideA;
  const float* Bb = B + (long)blockIdx.z * strideB;
  float*       Cb = C + (long)blockIdx.z * strideC;

  const float* arow = Ab + (long)(tileM + l15) * K;
  v8f acc = {};
  for (int k = 0; k < K; k += 4) {
    const int ka = k + 2 * half;       // this half-wave supplies K=ka, ka+1
    v2f a, b;
    a.x = arow[ka];
    a.y = arow[ka + 1];
    if (TRANS_B) {
      const float* bcol = Bb + (long)(tileN + l15) * K;   // B^T row = logical col
      b.x = bcol[ka];
      b.y = bcol[ka + 1];
    } else {
      b.x = Bb[(long)ka       * N + tileN + l15];
      b.y = Bb[(long)(ka + 1) * N + tileN + l15];
    }
    acc = __builtin_amdgcn_wmma_f32_16x16x4_f32(
        false, a, false, b, (short)0, acc, false, false);
  }

  const int col = tileN + l15;
  float bv = (MODE == 1 || MODE == 2) ? bias[col] : 0.f;
#pragma unroll
  for (int r = 0; r < 8; ++r) {
    int row = tileM + r + 8 * half;    // C/D layout: VGPR r -> M = r (+8 for hi half)
    float v = acc[r] + bv;
    if (MODE == 2 || MODE == 3) v = v > 0.f ? v : 0.f;
    Cb[(long)row * N + col] = v;
  }
}

// ---------------------------------------------------------------------------
// Elementwise / reduction kernels
// ---------------------------------------------------------------------------
__global__ void zero_acc(float* acc) { if (threadIdx.x < 4) acc[threadIdx.x] = 0.f; }

__global__ void concat_hx(const float* __restrict__ hs, const float* __restrict__ in,
                          float* __restrict__ out)
{
  long i = (long)blockIdx.x * blockDim.x + threadIdx.x;
  if (i >= (long)ROWS * 192) return;
  int r = (int)(i / 192), c = (int)(i % 192);
  out[i] = (c < HH) ? hs[(long)r * HH + c] : in[(long)r * F_IN + (c - HH)];
}

// mu/sigma/latent from gc, entropy partial sum (block reduce -> one atomic)
__global__ void latent_kernel(const float* __restrict__ gc, const float* __restrict__ eps,
                              float* __restrict__ mu, float* __restrict__ sigma,
                              float* __restrict__ latent, float* __restrict__ acc)
{
  int i = blockIdx.x * 256 + threadIdx.x;
  float logs = 0.f;
  if (i < ROWS * LL) {
    int row = i >> 6, l = i & 63;
    float m  = gc[(long)row * 128 + l];
    float vv = expf(gc[(long)row * 128 + 64 + l]);
    vv = vv < 0.002f ? 0.002f : vv;
    float sg = sqrtf(vv);
    mu[i] = m; sigma[i] = sg;
    latent[i] = m + sg * eps[i];
    logs = logf(sg);
  }
  __shared__ float sh[256];
  sh[threadIdx.x] = logs; __syncthreads();
  for (int s = 128; s > 0; s >>= 1) {
    if (threadIdx.x < s) sh[threadIdx.x] += sh[threadIdx.x + s];
    __syncthreads();
  }
  if (threadIdx.x == 0) atomicAdd(&acc[0], sh[0]);
}

// KL partial sum from inf head
__global__ void kl_kernel(const float* __restrict__ inf, const float* __restrict__ mu,
                          const float* __restrict__ sigma, float* __restrict__ acc)
{
  int i = blockIdx.x * 256 + threadIdx.x;
  float e = 0.f;
  if (i < ROWS * LL) {
    int row = i >> 6, l = i & 63;
    float m2 = inf[(long)row * 128 + l];
    float v2 = expf(inf[(long)row * 128 + 64 + l]);
    v2 = v2 < 0.002f ? 0.002f : v2;
    float s2 = sqrtf(v2);
    float s1 = sigma[i], m1 = mu[i];
    float d  = m1 - m2;
    e = logf(s2 / s1) + (s1 * s1 + d * d) / (2.f * v2) - 0.5f;
  }
  __shared__ float sh[256];
  sh[threadIdx.x] = e; __syncthreads();
  for (int s = 128; s > 0; s >>= 1) {
    if (threadIdx.x < s) sh[threadIdx.x] += sh[threadIdx.x + s];
    __syncthreads();
  }
  if (threadIdx.x == 0) atomicAdd(&acc[1], sh[0]);
}

// per-column mean / invstd over Mrows (BN axis=0); one block per column
__global__ void colstats(const float* __restrict__ X, int Mrows, int Ncols,
                         float* __restrict__ mean, float* __restrict__ invstd)
{
  int c = blockIdx.x;
  float s = 0.f, q = 0.f;
  for (int r = threadIdx.x; r < Mrows; r += 256) {
    float v = X[(long)r * Ncols + c]; s += v; q += v * v;
  }
  __shared__ float ss[256], sq[256];
  ss[threadIdx.x] = s; sq[threadIdx.x] = q; __syncthreads();
  for (int st = 128; st > 0; st >>= 1) {
    if (threadIdx.x < st) { ss[threadIdx.x] += ss[threadIdx.x + st];
                            sq[threadIdx.x] += sq[threadIdx.x + st]; }
    __syncthreads();
  }
  if (threadIdx.x == 0) {
    float m = ss[0] / (float)Mrows;
    float var = sq[0] / (float)Mrows - m * m;
    mean[c] = m; invstd[c] = rsqrtf(var + 1e-5f);
  }
}

__global__ void bn_leaky(float* __restrict__ X, const float* __restrict__ mean,
                         const float* __restrict__ invstd, const float* __restrict__ g,
                         const float* __restrict__ be, long total, int Ncols)
{
  long i = (long)blockIdx.x * blockDim.x + threadIdx.x;
  if (i >= total) return;
  int c = (int)(i % Ncols);
  X[i] = leakyf((X[i] - mean[c]) * invstd[c] * g[c] + be[c]);
}

// mi_cat[b, k*A+a]
__global__ void mi_kernel(const float* __restrict__ latent, const float* __restrict__ mu,
                          const float* __restrict__ sigma, float* __restrict__ mi)
{
  int i = blockIdx.x * 256 + threadIdx.x;
  if (i >= BS * AA * AA) return;
  int b = i >> 12, k = (i >> 6) & 63, a = i & 63;
  int src = (a - k - 1) & 63;
  const float* rl = latent + ((long)b * AA + src) * LL;
  const float* mm = mu     + ((long)b * AA + a)   * LL;
  const float* ssg= sigma  + ((long)b * AA + a)   * LL;
  float s = 0.f;
#pragma unroll 4
  for (int l = 0; l < LL; ++l) {
    float d  = (rl[l] - mm[l]) / ssg[l];
    float lp = -0.5f * d * d - logf(ssg[l]) - 0.5f * LOG2PI;
    float v  = lp + 13.9f; v = v < -13.9f ? -13.9f : v;
    s += v;
  }
  mi[i] = s / (float)LL;
}

// dh BN stats over (b,a) per (k,n); rotation in `a` is a full cycle -> iterate directly
__global__ void dh_stats(const float* __restrict__ T1, const float* __restrict__ T2,
                         float* __restrict__ dmean, float* __restrict__ dinv)
{
  int k = blockIdx.x, n = threadIdx.x;
  float s = 0.f, q = 0.f;
  for (int b = 0; b < BS; ++b) {
    float t1 = T1[((long)b * AA + k) * NNH + n];
    const float* t2p = T2 + (long)b * AA * NNH + n;
    for (int a = 0; a < AA; ++a) {
      float v = t1 + t2p[(long)a * NNH];
      s += v; q += v * v;
    }
  }
  const float cnt = (float)(BS * AA);
  float m = s / cnt;
  float var = q / cnt - m * m;
  dmean[k * NNH + n] = m;
  dinv [k * NNH + n] = rsqrtf(var + 1e-5f);
}

// dis_cat[b, k*A+a] = |sum_n leaky(norm(T1+T2)) * w2[n] + b2|
__global__ void dis_kernel(const float* __restrict__ T1, const float* __restrict__ T2,
                           const float* __restrict__ dmean, const float* __restrict__ dinv,
                           const float* __restrict__ g1, const float* __restrict__ be1,
                           const float* __restrict__ w2, const float* __restrict__ b2,
                           float* __restrict__ dis)
{
  int i = blockIdx.x * 256 + threadIdx.x;
  if (i >= BS * AA * AA) return;
  int b = i >> 12, k = (i >> 6) & 63, a = i & 63;
  int src = (a - k - 1) & 63;
  const float* t1 = T1 + ((long)b * AA + k)   * NNH;
  const float* t2 = T2 + ((long)b * AA + src) * NNH;
  const float* dm = dmean + k * NNH;
  const float* di = dinv  + k * NNH;
  float acc = 0.f;
#pragma unroll 4
  for (int n = 0; n < NNH; ++n) {
    float v = (t1[n] + t2[n] - dm[n]) * di[n] * g1[n] + be1[n];
    acc += leakyf(v) * w2[n];
  }
  dis[i] = fabsf(acc + b2[0]);
}

__global__ void row_minmax(const float* __restrict__ X, float* __restrict__ rmin,
                           float* __restrict__ rmax)
{
  int row = blockIdx.x;
  const float* p = X + (long)row * (AA * AA);
  float lo = 3.4e38f, hi = -3.4e38f;
  for (int j = threadIdx.x; j < AA * AA; j += 256) {
    float v = p[j]; lo = fminf(lo, v); hi = fmaxf(hi, v);
  }
  __shared__ float smin[256], smax[256];
  smin[threadIdx.x] = lo; smax[threadIdx.x] = hi; __syncthreads();
  for (int s = 128; s > 0; s >>= 1) {
    if (threadIdx.x < s) { smin[threadIdx.x] = fminf(smin[threadIdx.x], smin[threadIdx.x + s]);
                           smax[threadIdx.x] = fmaxf(smax[threadIdx.x], smax[threadIdx.x + s]); }
    __syncthreads();
  }
  if (threadIdx.x == 0) { rmin[row] = smin[0]; rmax[row] = smax[0]; }
}

__global__ void final_dis(const float* __restrict__ mi, const float* __restrict__ di,
                          const float* __restrict__ mimin, const float* __restrict__ mimax,
                          const float* __restrict__ dimin, const float* __restrict__ dimax,
                          float* __restrict__ acc)
{
  int b = blockIdx.x;
  float mlo = mimin[b], mrng = mimax[b] - mlo + 1e-12f;
  float dlo = dimin[b], drng = dimax[b] - dlo + 1e-12f;
  float cs = 0.f, ns = 0.f;
  for (int j = threadIdx.x; j < AA * AA; j += 256) {
    float mn = (mi[(long)b * AA * AA + j] - mlo) / mrng;
    float dn = (di[(long)b * AA * AA + j] - dlo) / drng;
    float c = mn + dn; c = c > 1.f ? 1.f : c;
    cs += c; ns += fabsf(dn);
  }
  __shared__ float sc[256], sn[256];
  sc[threadIdx.x] = cs; sn[threadIdx.x] = ns; __syncthreads();
  for (int s = 128; s > 0; s >>= 1) {
    if (threadIdx.x < s) { sc[threadIdx.x] += sc[threadIdx.x + s];
                           sn[threadIdx.x] += sn[threadIdx.x + s]; }
    __syncthreads();
  }
  if (threadIdx.x == 0) { atomicAdd(&acc[2], sc[0]); atomicAdd(&acc[3], sn[0]); }
}

__global__ void gru_kernel(const float* __restrict__ gi, const float* __restrict__ gh,
                           const float* __restrict__ h_in, float* __restrict__ hw,
                           float* __restrict__ hout)
{
  int i = blockIdx.x * 256 + threadIdx.x;
  if (i >= ROWS * HH) return;
  int r = i >> 6, c = i & 63;
  float ir = gi[(long)r * 192 + c],       hr = gh[(long)r * 192 + c];
  float iz = gi[(long)r * 192 + 64 + c],  hz = gh[(long)r * 192 + 64 + c];
  float in_= gi[(long)r * 192 + 128 + c], hn = gh[(long)r * 192 + 128 + c];
  float rr = 1.f / (1.f + expf(-(ir + hr)));
  float zz = 1.f / (1.f + expf(-(iz + hz)));
  float nn = tanhf(in_ + rr * hn);
  float h  = (1.f - zz) * nn + zz * h_in[i];
  hw[i] = h; hout[i] = h;
}

__global__ void q_kernel(const float* __restrict__ h, const float* __restrict__ fc2w,
                         const float* __restrict__ fc2b, float* __restrict__ q)
{
  int i = blockIdx.x * 256 + threadIdx.x;
  if (i >= ROWS * NACT) return;
  int b = i >> 4, a = i & 15;
  const float* w  = fc2w + (long)b * (HH * NACT);
  const float* hb = h + (long)b * HH;
  float s = 0.f;
#pragma unroll 8
  for (int k = 0; k < HH; ++k) s += hb[k] * w[k * NACT + a];
  q[i] = s + fc2b[i];
}

__global__ void finalize(const float* __restrict__ acc, float* __restrict__ out)
{
  if (threadIdx.x | blockIdx.x) return;
  float ent = 64.f * (0.5f + 0.5f * LOG2PI) + acc[0] / (float)ROWS;
  float kl  = acc[1] / (float)ROWS;
  float loss = ent * 0.01f + kl * 0.0001f;
  loss = loss > 2000.f ? 2000.f : loss;
  float ce = log1pf(expf(loss));
  float dis_loss = -acc[2] / (float)(BS * AA);
  float dis_norm =  acc[3] / (float)(BS * AA);
  float c_dis = (dis_norm + 1.0f * dis_loss) / (float)AA * 0.001f;
  out[655360] = ce + c_dis;   // loss_total
  out[655361] = c_dis;        // c_dis_loss
  out[655362] = ce;           // ce_loss
}

// ---------------------------------------------------------------------------
// Launcher
// ---------------------------------------------------------------------------
extern "C" void kernel_launch(void* const* d_in, const int* in_sizes, int n_in,
                              void* d_out, int out_size, void* d_ws, size_t ws_size,
                              hipStream_t stream)
{
  const float* inputs  = (const float*)d_in[0];
  const float* hidden  = (const float*)d_in[1];
  const float* adj     = (const float*)d_in[2];
  const float* eps     = (const float*)d_in[3];
  const float* gc_w    = (const float*)d_in[4];
  const float* inf_w1  = (const float*)d_in[5];
  const float* inf_b1  = (const float*)d_in[6];
  const float* inf_g1  = (const float*)d_in[7];
  const float* inf_be1 = (const float*)d_in[8];
  const float* inf_w2  = (const float*)d_in[9];
  const float* inf_b2  = (const float*)d_in[10];
  const float* lat_w   = (const float*)d_in[11];
  const float* lat_b   = (const float*)d_in[12];
  const float* lat_g   = (const float*)d_in[13];
  const float* lat_be  = (const float*)d_in[14];
  const float* fc1_w   = (const float*)d_in[15];
  const float* fc1_b   = (const float*)d_in[16];
  const float* gru_wih = (const float*)d_in[17];
  const float* gru_whh = (const float*)d_in[18];
  const float* gru_bih = (const float*)d_in[19];
  const float* gru_bhh = (const float*)d_in[20];
  const float* w_nn_w  = (const float*)d_in[21];
  const float* w_nn_b  = (const float*)d_in[22];
  const float* b_nn_w  = (const float*)d_in[23];
  const float* b_nn_b  = (const float*)d_in[24];
  const float* dis_w1  = (const float*)d_in[25];
  const float* dis_g1  = (const float*)d_in[27];
  const float* dis_be1 = (const float*)d_in[28];
  const float* dis_w2  = (const float*)d_in[29];
  const float* dis_b2  = (const float*)d_in[30];

  float* ws  = (float*)d_ws;
  float* out = (float*)d_out;
  float* q_out = out;                    // 8192*16
  float* h_out = out + 131072;           // 8192*64

  float* acc = ws + WS_ACC;

  zero_acc<<<1, 32, 0, stream>>>(acc);
  concat_hx<<<(ROWS * 192 + 255) / 256, 256, 0, stream>>>(hidden, inputs, ws + WS_INFIN);

  // support = inputs @ gc_w    (8192x128 @ 128x128)
  gemm_wmma<0, 0><<<dim3(128 / 16, ROWS / 64, 1), 128, 0, stream>>>(
      inputs, gc_w, nullptr, ws + WS_SUPPORT, ROWS, 128, 128, 0, 0, 0);
  // gc = relu(adj @ support[b])  batched over BS
  gemm_wmma<0, 3><<<dim3(128 / 16, 1, BS), 128, 0, stream>>>(
      adj, ws + WS_SUPPORT, nullptr, ws + WS_GC, AA, 128, AA,
      0, (long)AA * 128, (long)AA * 128);
  // mu/sigma/latent + entropy sum
  latent_kernel<<<(ROWS * LL) / 256, 256, 0, stream>>>(
      ws + WS_GC, eps, ws + WS_MU, ws + WS_SIGMA, ws + WS_LATENT, acc);

  // inference MLP
  gemm_wmma<0, 1><<<dim3(NNH / 16, ROWS / 64, 1), 128, 0, stream>>>(
      ws + WS_INFIN, inf_w1, inf_b1, ws + WS_INFH, ROWS, NNH, 192, 0, 0, 0);
  colstats<<<NNH, 256, 0, stream>>>(ws + WS_INFH, ROWS, NNH, ws + WS_BNMEAN, ws + WS_BNINV);
  bn_leaky<<<(ROWS * NNH + 255) / 256, 256, 0, stream>>>(
      ws + WS_INFH, ws + WS_BNMEAN, ws + WS_BNINV, inf_g1, inf_be1, (long)ROWS * NNH, NNH);
  gemm_wmma<0, 1><<<dim3(128 / 16, ROWS / 64, 1), 128, 0, stream>>>(
      ws + WS_INFH, inf_w2, inf_b2, ws + WS_INFOUT, ROWS, 128, NNH, 0, 0, 0);
  kl_kernel<<<(ROWS * LL) / 256, 256, 0, stream>>>(
      ws + WS_INFOUT, ws + WS_MU, ws + WS_SIGMA, acc);

  // discriminator: T1 = latent @ W1[:64], T2 = latent @ W1[64:]
  gemm_wmma<0, 0><<<dim3(NNH / 16, ROWS / 64, 1), 128, 0, stream>>>(
      ws + WS_LATENT, dis_w1, nullptr, ws + WS_T1, ROWS, NNH, LL, 0, 0, 0);
  gemm_wmma<0, 0><<<dim3(NNH / 16, ROWS / 64, 1), 128, 0, stream>>>(
      ws + WS_LATENT, dis_w1 + (long)LL * NNH, nullptr, ws + WS_T2, ROWS, NNH, LL, 0, 0, 0);
  dh_stats<<<AA, NNH, 0, stream>>>(ws + WS_T1, ws + WS_T2, ws + WS_DHMEAN, ws + WS_DHINV);
  mi_kernel<<<(BS * AA * AA) / 256, 256, 0, stream>>>(
      ws + WS_LATENT, ws + WS_MU, ws + WS_SIGMA, ws + WS_MICAT);
  dis_kernel<<<(BS * AA * AA) / 256, 256, 0, stream>>>(
      ws + WS_T1, ws + WS_T2, ws + WS_DHMEAN, ws + WS_DHINV,
      dis_g1, dis_be1, dis_w2, dis_b2, ws + WS_DISCAT);
  row_minmax<<<BS, 256, 0, stream>>>(ws + WS_MICAT, ws + WS_MIMIN, ws + WS_MIMAX);
  row_minmax<<<BS, 256, 0, stream>>>(ws + WS_DISCAT, ws + WS_DIMIN, ws + WS_DIMAX);
  final_dis<<<BS, 256, 0, stream>>>(ws + WS_MICAT, ws + WS_DISCAT,
      ws + WS_MIMIN, ws + WS_MIMAX, ws + WS_DIMIN, ws + WS_DIMAX, acc);

  // hypernetwork: z = leaky(BN(latent @ lat_w + lat_b))
  gemm_wmma<0, 1><<<dim3(NNH / 16, ROWS / 64, 1), 128, 0, stream>>>(
      ws + WS_LATENT, lat_w, lat_b, ws + WS_Z, ROWS, NNH, LL, 0, 0, 0);
  colstats<<<NNH, 256, 0, stream>>>(ws + WS_Z, ROWS, NNH, ws + WS_BNMEAN, ws + WS_BNINV);
  bn_leaky<<<(ROWS * NNH + 255) / 256, 256, 0, stream>>>(
      ws + WS_Z, ws + WS_BNMEAN, ws + WS_BNINV, lat_g, lat_be, (long)ROWS * NNH, NNH);
  gemm_wmma<0, 1><<<dim3(1024 / 16, ROWS / 64, 1), 128, 0, stream>>>(
      ws + WS_Z, w_nn_w, w_nn_b, ws + WS_FC2W, ROWS, 1024, NNH, 0, 0, 0);
  gemm_wmma<0, 1><<<dim3(NACT / 16, ROWS / 64, 1), 128, 0, stream>>>(
      ws + WS_Z, b_nn_w, b_nn_b, ws + WS_FC2B, ROWS, NACT, NNH, 0, 0, 0);

  // GRU path
  gemm_wmma<0, 2><<<dim3(HH / 16, ROWS / 64, 1), 128, 0, stream>>>(
      inputs, fc1_w, fc1_b, ws + WS_X, ROWS, HH, F_IN, 0, 0, 0);
  gemm_wmma<1, 1><<<dim3(192 / 16, ROWS / 64, 1), 128, 0, stream>>>(
      ws + WS_X, gru_wih, gru_bih, ws + WS_GI, ROWS, 192, HH, 0, 0, 0);
  gemm_wmma<1, 1><<<dim3(192 / 16, ROWS / 64, 1), 128, 0, stream>>>(
      hidden, gru_whh, gru_bhh, ws + WS_GH, ROWS, 192, HH, 0, 0, 0);
  gru_kernel<<<(ROWS * HH) / 256, 256, 0, stream>>>(
      ws + WS_GI, ws + WS_GH, hidden, ws + WS_H, h_out);
  q_kernel<<<(ROWS * NACT) / 256, 256, 0, stream>>>(
      ws + WS_H, ws + WS_FC2W, ws + WS_FC2B, q_out);

  finalize<<<1, 32, 0, stream>>>(acc, out);
}